// MyGNN_87136296501830
// MI455X (gfx1250) — compile-verified
//
#include <hip/hip_runtime.h>
#include <hip/hip_bf16.h>

#define N_NODES   50000
#define N_EDGES   800000
#define CH        128
#define OUT_CH    12
#define NUM_GRAPHS 512

typedef __attribute__((ext_vector_type(2))) float v2f;
typedef __attribute__((ext_vector_type(8))) float v8f;

// ---------------- degree / normalization ----------------
__global__ void deg_init_kernel(float* deg) {
    int i = blockIdx.x * blockDim.x + threadIdx.x;
    if (i < N_NODES) deg[i] = 1.0f;   // self-loop contributes 1
}

__global__ void deg_edges_kernel(const int* __restrict__ dst, float* deg) {
    int e = blockIdx.x * blockDim.x + threadIdx.x;
    if (e < N_EDGES) atomicAdd(&deg[dst[e]], 1.0f);
}

__global__ void dinv_kernel(float* deg) {
    int i = blockIdx.x * blockDim.x + threadIdx.x;
    if (i < N_NODES) deg[i] = rsqrtf(deg[i]);   // deg >= 1 always
}

// ---------------- WMMA GEMM: H = (relu? X) @ W,  [N_NODES x CH] @ [CH x CH] ----------------
// One wave computes a full 16x128 strip: 8 fp32 accumulators, so each A fragment
// (loaded once) feeds 8 v_wmma_f32_16x16x4_f32 issues. X is read exactly once.
// Tail tiles: clamp A row (loads in-bounds, EXEC all-1 through WMMA), guard stores.
template <int RELU>
__global__ __launch_bounds__(256) void gcn_gemm_wmma(
    const float* __restrict__ X, const float* __restrict__ W,
    float* __restrict__ H)
{
    const int lane = threadIdx.x & 31;
    const int wave = threadIdx.x >> 5;
    const int l16  = lane & 15;
    const int half = lane >> 4;              // 0: lanes 0-15, 1: lanes 16-31
    const int m0   = (blockIdx.x * 8 + wave) * 16;
    int arow = m0 + l16;
    if (arow > N_NODES - 1) arow = N_NODES - 1;

    v8f acc[8];
    #pragma unroll
    for (int j = 0; j < 8; ++j) acc[j] = (v8f){};

    for (int kb = 0; kb < CH; kb += 4) {
        const int k = kb + 2 * half;         // this lane's K pair
        v2f a;
        a.x = X[(size_t)arow * CH + k + 0];
        a.y = X[(size_t)arow * CH + k + 1];
        if (RELU) { a.x = fmaxf(a.x, 0.0f); a.y = fmaxf(a.y, 0.0f); }
        #pragma unroll
        for (int j = 0; j < 8; ++j) {
            v2f b;
            b.x = W[(k + 0) * CH + j * 16 + l16];
            b.y = W[(k + 1) * CH + j * 16 + l16];
            acc[j] = __builtin_amdgcn_wmma_f32_16x16x4_f32(
                false, a, false, b, (short)0, acc[j], false, false);
        }
    }

    #pragma unroll
    for (int r = 0; r < 8; ++r) {
        const int orow = m0 + r + 8 * half;
        if (orow < N_NODES) {
            #pragma unroll
            for (int j = 0; j < 8; ++j)
                H[(size_t)orow * CH + j * 16 + l16] = acc[j][r];
        }
    }
}

// ---------------- out[v] = dinv[v]^2 * h[v] + bias  (self-loop + bias init) ----------------
__global__ __launch_bounds__(256) void node_init_kernel(
    const float* __restrict__ h, const float* __restrict__ dinv,
    const float* __restrict__ bias, float* __restrict__ out)
{
    const int v    = blockIdx.x * (blockDim.x >> 5) + (threadIdx.x >> 5);
    const int lane = threadIdx.x & 31;
    if (v >= N_NODES) return;
    const float di = dinv[v];
    const float c  = di * di;
    const float4 hv = *(const float4*)(h + (size_t)v * CH + lane * 4);
    const float4 bv = *(const float4*)(bias + lane * 4);
    float4 o;
    o.x = c * hv.x + bv.x;  o.y = c * hv.y + bv.y;
    o.z = c * hv.z + bv.z;  o.w = c * hv.w + bv.w;
    *(float4*)(out + (size_t)v * CH + lane * 4) = o;
}

// ---------------- edge scatter: out[d] += dinv[s]*dinv[d]*h[s]  (one wave per edge) ---------
__global__ __launch_bounds__(256) void edge_agg_kernel(
    const int* __restrict__ src, const int* __restrict__ dst,
    const float* __restrict__ dinv,
    const float* __restrict__ h, float* __restrict__ out)
{
    const int e    = blockIdx.x * (blockDim.x >> 5) + (threadIdx.x >> 5);
    const int lane = threadIdx.x & 31;
    if (e >= N_EDGES) return;
    const int s = src[e], d = dst[e];
    const float norm = dinv[s] * dinv[d];
    const float4 hv = *(const float4*)(h + (size_t)s * CH + lane * 4);
    float* o = out + (size_t)d * CH + lane * 4;
    atomicAdd(o + 0, norm * hv.x);
    atomicAdd(o + 1, norm * hv.y);
    atomicAdd(o + 2, norm * hv.z);
    atomicAdd(o + 3, norm * hv.w);
}

// ---------------- pooling ----------------
__global__ void fzero_kernel(float* p, int n) {
    int i = blockIdx.x * blockDim.x + threadIdx.x;
    if (i < n) p[i] = 0.0f;
}

__global__ __launch_bounds__(256) void pool_kernel(
    const float* __restrict__ h, const int* __restrict__ batch,
    float* __restrict__ sums, float* __restrict__ cnt)
{
    const int v    = blockIdx.x * (blockDim.x >> 5) + (threadIdx.x >> 5);
    const int lane = threadIdx.x & 31;
    if (v >= N_NODES) return;
    const int g = batch[v];
    const float4 hv = *(const float4*)(h + (size_t)v * CH + lane * 4);
    float* s = sums + (size_t)g * CH + lane * 4;
    atomicAdd(s + 0, hv.x);
    atomicAdd(s + 1, hv.y);
    atomicAdd(s + 2, hv.z);
    atomicAdd(s + 3, hv.w);
    if (lane == 0) atomicAdd(&cnt[g], 1.0f);
}

// ---------------- classifier: out[g] = (sums[g]/max(cnt,1)) @ Wc + bc ----------------
__global__ void classify_kernel(
    const float* __restrict__ sums, const float* __restrict__ cnt,
    const float* __restrict__ Wc, const float* __restrict__ bc,
    float* __restrict__ out)
{
    const int g  = blockIdx.x;
    const int oc = threadIdx.x;
    if (oc >= OUT_CH) return;
    const float inv = 1.0f / fmaxf(cnt[g], 1.0f);
    float acc = bc[oc];
    for (int c = 0; c < CH; ++c)
        acc = fmaf(sums[(size_t)g * CH + c] * inv, Wc[c * OUT_CH + oc], acc);
    out[g * OUT_CH + oc] = acc;
}

// ---------------- host launch ----------------
extern "C" void kernel_launch(void* const* d_in, const int* in_sizes, int n_in,
                              void* d_out, int out_size, void* d_ws, size_t ws_size,
                              hipStream_t stream)
{
    (void)in_sizes; (void)n_in; (void)out_size; (void)ws_size;

    const float* x   = (const float*)d_in[0];
    const int*   ei  = (const int*)d_in[1];
    const int*   bat = (const int*)d_in[2];
    const float* W0  = (const float*)d_in[3];
    const float* b0  = (const float*)d_in[4];
    const float* W1  = (const float*)d_in[5];
    const float* b1  = (const float*)d_in[6];
    const float* W2  = (const float*)d_in[7];
    const float* b2  = (const float*)d_in[8];
    const float* Wc  = (const float*)d_in[9];
    const float* bc  = (const float*)d_in[10];
    const int* src = ei;
    const int* dst = ei + N_EDGES;

    char* ws = (char*)d_ws;
    size_t off = 0;
    auto carve = [&](size_t bytes) -> void* {
        void* p = ws + off;
        off = (off + bytes + 255) & ~(size_t)255;
        return p;
    };
    float* dinv = (float*)carve((size_t)N_NODES * 4);
    float* hA   = (float*)carve((size_t)N_NODES * CH * 4);
    float* hB   = (float*)carve((size_t)N_NODES * CH * 4);
    float* sums = (float*)carve((size_t)NUM_GRAPHS * CH * 4);
    float* cnt  = (float*)carve((size_t)NUM_GRAPHS * 4);

    const int TB = 256;
    const int N_MTILES   = (N_NODES + 15) / 16;          // 3125
    const int GRID_GEMM  = (N_MTILES + 7) / 8;           // 391 (8 strips/block)
    const int GRID_NODEW = (N_NODES + 7) / 8;            // 8 waves/block
    const int GRID_EDGEW = (N_EDGES + 7) / 8;

    // degree -> dinv (recomputed every call; ws is not trusted across calls)
    deg_init_kernel<<<(N_NODES + TB - 1) / TB, TB, 0, stream>>>(dinv);
    deg_edges_kernel<<<(N_EDGES + TB - 1) / TB, TB, 0, stream>>>(dst, dinv);
    dinv_kernel<<<(N_NODES + TB - 1) / TB, TB, 0, stream>>>(dinv);

    // layer 0 (no input relu)
    gcn_gemm_wmma<0><<<GRID_GEMM, TB, 0, stream>>>(x, W0, hA);
    node_init_kernel<<<GRID_NODEW, TB, 0, stream>>>(hA, dinv, b0, hB);
    edge_agg_kernel<<<GRID_EDGEW, TB, 0, stream>>>(src, dst, dinv, hA, hB);

    // layer 1 (relu fused into A-load)
    gcn_gemm_wmma<1><<<GRID_GEMM, TB, 0, stream>>>(hB, W1, hA);
    node_init_kernel<<<GRID_NODEW, TB, 0, stream>>>(hA, dinv, b1, hB);
    edge_agg_kernel<<<GRID_EDGEW, TB, 0, stream>>>(src, dst, dinv, hA, hB);

    // layer 2
    gcn_gemm_wmma<1><<<GRID_GEMM, TB, 0, stream>>>(hB, W2, hA);
    node_init_kernel<<<GRID_NODEW, TB, 0, stream>>>(hA, dinv, b2, hB);
    edge_agg_kernel<<<GRID_EDGEW, TB, 0, stream>>>(src, dst, dinv, hA, hB);

    // mean pool + classifier
    fzero_kernel<<<(NUM_GRAPHS * CH + TB - 1) / TB, TB, 0, stream>>>(sums, NUM_GRAPHS * CH);
    fzero_kernel<<<(NUM_GRAPHS + TB - 1) / TB, TB, 0, stream>>>(cnt, NUM_GRAPHS);
    pool_kernel<<<GRID_NODEW, TB, 0, stream>>>(hB, bat, sums, cnt);
    classify_kernel<<<NUM_GRAPHS, 32, 0, stream>>>(sums, cnt, Wc, bc, (float*)d_out);
}